// MultiHeadAttention_50964081934967
// MI455X (gfx1250) — compile-verified
//
#include <hip/hip_runtime.h>

// ---------------------------------------------------------------------------
// MHA forward for MI455X (gfx1250, wave32) using V_WMMA_F32_16X16X4_F32.
// Outputs: out [B,S,E] followed by attn [B,H,S,S] (flat, fp32).
// ---------------------------------------------------------------------------

typedef __attribute__((ext_vector_type(2))) float v2f;
typedef __attribute__((ext_vector_type(8))) float v8f;

#define WMMA_F32(a, b, c) \
  __builtin_amdgcn_wmma_f32_16x16x4_f32(false, (a), false, (b), (short)0, (c), false, false)

namespace {
constexpr int kB = 2;
constexpr int kS = 2048;
constexpr int kE = 1024;
constexpr int kH = 16;
constexpr int kD = 64;
constexpr int kMaxRel = 512;
constexpr int kM = kB * kS;          // 4096 rows for all projections
constexpr int kSStride = 2052;       // padded LDS score row stride (4 mod 64 -> bank-conflict free)
constexpr int kQStride = 68;         // padded LDS q-tile row stride
}  // namespace

// ---------------------------------------------------------------------------
// C[M,N] = A[M,K] @ W[N,K]^T + bias[N]      (torch Linear)
// Block: 256 threads = 8 waves; wave computes a 32x32 tile; block tile 64x128.
// ---------------------------------------------------------------------------
__global__ __launch_bounds__(256) void gemm_bias_wmma(
    const float* __restrict__ A, const float* __restrict__ W,
    const float* __restrict__ bias, float* __restrict__ C,
    int M, int N, int K) {
  const int lane = threadIdx.x & 31;
  const int wave = threadIdx.x >> 5;
  const int half = lane >> 4;   // selects k-offset {0,2}
  const int lr   = lane & 15;   // row (A) / row (W) within 16
  const int wm = wave >> 2;     // 0..1
  const int wn = wave & 3;      // 0..3
  const int m0 = blockIdx.x * 64 + wm * 32;
  const int n0 = blockIdx.y * 128 + wn * 32;

  v8f acc00 = {}, acc01 = {}, acc10 = {}, acc11 = {};
  const float* Ar0 = A + (size_t)(m0 + lr) * K + half * 2;
  const float* Ar1 = A + (size_t)(m0 + 16 + lr) * K + half * 2;
  const float* Wr0 = W + (size_t)(n0 + lr) * K + half * 2;
  const float* Wr1 = W + (size_t)(n0 + 16 + lr) * K + half * 2;

  for (int k = 0; k < K; k += 4) {
    v2f a0 = *(const v2f*)(Ar0 + k);
    v2f a1 = *(const v2f*)(Ar1 + k);
    v2f b0 = *(const v2f*)(Wr0 + k);
    v2f b1 = *(const v2f*)(Wr1 + k);
    acc00 = WMMA_F32(a0, b0, acc00);
    acc01 = WMMA_F32(a0, b1, acc01);
    acc10 = WMMA_F32(a1, b0, acc10);
    acc11 = WMMA_F32(a1, b1, acc11);
  }

  const float bv0 = bias[n0 + lr];
  const float bv1 = bias[n0 + 16 + lr];
#pragma unroll
  for (int r = 0; r < 8; ++r) {
    const int m = r + half * 8;  // C layout: vgpr r, lanes 0-15 -> M=r, lanes 16-31 -> M=8+r
    C[(size_t)(m0 + m) * N + n0 + lr]           = acc00[r] + bv0;
    C[(size_t)(m0 + m) * N + n0 + 16 + lr]      = acc01[r] + bv1;
    C[(size_t)(m0 + 16 + m) * N + n0 + lr]      = acc10[r] + bv0;
    C[(size_t)(m0 + 16 + m) * N + n0 + 16 + lr] = acc11[r] + bv1;
  }
}

// ---------------------------------------------------------------------------
// Fused scores -> +relbias -> causal mask -> softmax -> attn write -> attn@V.
// One block per (b, h, 16-row query tile); 8 waves each own 256 key columns.
// Dynamic LDS layout (floats):
//   s    [16][2052]  scores / probs stripe (later reused for AV partials)
//   qt   [16][68]    staged Q tile
//   bi   [1025]      rel-bias table column for this head
//   red  [256]       softmax partial reductions
//   stat [32]        row max (0..15) and row 1/sum (16..31)
// ---------------------------------------------------------------------------
__global__ __launch_bounds__(256) void attn_softmax_av(
    const float* __restrict__ Q, const float* __restrict__ Km,
    const float* __restrict__ V, const float* __restrict__ table,
    float* __restrict__ attn, float* __restrict__ Ahead) {
  extern __shared__ float lds[];
  float* s    = lds;
  float* qt   = s + 16 * kSStride;
  float* bi   = qt + 16 * kQStride;
  float* red  = bi + (2 * kMaxRel + 1) + 3;
  float* stat = red + 256;

  const int tid  = threadIdx.x;
  const int lane = tid & 31;
  const int wave = tid >> 5;
  const int half = lane >> 4;
  const int lr   = lane & 15;

  const int qtile = blockIdx.x & 127;        // S/16 = 128
  const int h     = (blockIdx.x >> 7) & 15;
  const int b     = blockIdx.x >> 11;
  const int q0    = qtile * 16;

  // ---- stage Q tile and bias column ----
  for (int i = tid; i < 16 * kD; i += 256) {
    const int r = i >> 6, c = i & 63;
    qt[r * kQStride + c] = Q[((size_t)(b * kS + q0 + r)) * kE + h * kD + c];
  }
  for (int i = tid; i < 2 * kMaxRel + 1; i += 256) bi[i] = table[(size_t)i * kH + h];
  __syncthreads();

  // ---- phase 1: scores for this wave's 256 key columns ----
  v2f afrag[16];
#pragma unroll
  for (int kk = 0; kk < 16; ++kk)
    afrag[kk] = *(const v2f*)(qt + lr * kQStride + kk * 4 + half * 2);

  const int c0w = wave * 256;
  for (int nt = 0; nt < 16; ++nt) {
    const int c0 = c0w + nt * 16;
    const float* krow = Km + ((size_t)(b * kS + c0 + lr)) * kE + h * kD + half * 2;
    v8f c = {};
#pragma unroll
    for (int kk = 0; kk < 16; ++kk) {
      v2f bfrag = *(const v2f*)(krow + kk * 4);
      c = WMMA_F32(afrag[kk], bfrag, c);
    }
    const int ncol = c0 + lr;
#pragma unroll
    for (int r = 0; r < 8; ++r) {
      const int m  = r + half * 8;
      const int mq = q0 + m;
      float sc = c[r] * 0.125f;  // 1/sqrt(64)
      int rel = mq - ncol;
      rel = rel < -kMaxRel ? -kMaxRel : (rel > kMaxRel ? kMaxRel : rel);
      sc += bi[rel + kMaxRel];
      if (ncol > mq) sc = -__builtin_inff();
      s[m * kSStride + ncol] = sc;
    }
  }
  __syncthreads();

  // ---- phase 2: softmax over full 2048 columns; thread = (row, 16-lane slice)
  const int row = tid >> 4;
  const int lc  = tid & 15;
  float* srow = s + row * kSStride;

  float pmax = -__builtin_inff();
  for (int i = 0; i < 128; ++i) pmax = fmaxf(pmax, srow[lc + i * 16]);
  red[tid] = pmax;
  __syncthreads();
  if (lc == 0) {
    float m = red[row * 16];
    for (int j = 1; j < 16; ++j) m = fmaxf(m, red[row * 16 + j]);
    stat[row] = m;
  }
  __syncthreads();
  const float rmax = stat[row];
  float psum = 0.f;
  for (int i = 0; i < 128; ++i) {
    const float v = __expf(srow[lc + i * 16] - rmax);
    srow[lc + i * 16] = v;
    psum += v;
  }
  red[tid] = psum;
  __syncthreads();
  if (lc == 0) {
    float t = 0.f;
    for (int j = 0; j < 16; ++j) t += red[row * 16 + j];
    stat[16 + row] = 1.0f / t;
  }
  __syncthreads();
  const float inv = stat[16 + row];
  float* arow = attn + ((size_t)(b * kH + h) * kS + q0 + row) * kS;
  for (int i = 0; i < 128; ++i) {
    const float v = srow[lc + i * 16] * inv;
    srow[lc + i * 16] = v;
    arow[lc + i * 16] = v;  // attn probabilities: written to HBM exactly once
  }
  __syncthreads();

  // ---- phase 3: attn @ V, K-split across 8 waves, WMMA out of LDS ----
  v8f o0 = {}, o1 = {}, o2 = {}, o3 = {};
  const float* vbase = V + ((size_t)(b * kS + c0w)) * kE + h * kD;
  for (int kk = 0; kk < 64; ++kk) {
    v2f a = *(const v2f*)(s + lr * kSStride + c0w + kk * 4 + half * 2);
    const int k0 = kk * 4 + half * 2;
    const float* vp = vbase + (size_t)k0 * kE;
    v2f b0 = {vp[lr],      vp[kE + lr]};
    v2f b1 = {vp[16 + lr], vp[kE + 16 + lr]};
    v2f b2 = {vp[32 + lr], vp[kE + 32 + lr]};
    v2f b3 = {vp[48 + lr], vp[kE + 48 + lr]};
    o0 = WMMA_F32(a, b0, o0);
    o1 = WMMA_F32(a, b1, o1);
    o2 = WMMA_F32(a, b2, o2);
    o3 = WMMA_F32(a, b3, o3);
  }
  __syncthreads();  // all waves finished reading s; reuse it for partials

  float* part = s;  // [8][16][64]
#pragma unroll
  for (int r = 0; r < 8; ++r) {
    const int m = r + half * 8;
    part[((size_t)wave * 16 + m) * 64 + lr]      = o0[r];
    part[((size_t)wave * 16 + m) * 64 + 16 + lr] = o1[r];
    part[((size_t)wave * 16 + m) * 64 + 32 + lr] = o2[r];
    part[((size_t)wave * 16 + m) * 64 + 48 + lr] = o3[r];
  }
  __syncthreads();

  for (int i = tid; i < 16 * 64; i += 256) {
    const int m = i >> 6, n = i & 63;
    float acc = 0.f;
#pragma unroll
    for (int w = 0; w < 8; ++w) acc += part[((size_t)w * 16 + m) * 64 + n];
    Ahead[((size_t)(b * kS + q0 + m)) * kE + h * kD + n] = acc;
  }
}

// ---------------------------------------------------------------------------
extern "C" void kernel_launch(void* const* d_in, const int* in_sizes, int n_in,
                              void* d_out, int out_size, void* d_ws, size_t ws_size,
                              hipStream_t stream) {
  (void)in_sizes; (void)n_in; (void)out_size; (void)ws_size;

  const float* query = (const float*)d_in[0];
  const float* key   = (const float*)d_in[1];
  const float* value = (const float*)d_in[2];
  const float* Wq    = (const float*)d_in[3];
  const float* bq    = (const float*)d_in[4];
  const float* Wk    = (const float*)d_in[5];
  const float* bk    = (const float*)d_in[6];
  const float* Wv    = (const float*)d_in[7];
  const float* bv    = (const float*)d_in[8];
  const float* Wo    = (const float*)d_in[9];
  const float* bo    = (const float*)d_in[10];
  const float* tbl   = (const float*)d_in[11];

  float* out  = (float*)d_out;                       // [B,S,E]
  float* attn = out + (size_t)kB * kS * kE;          // [B,H,S,S]

  const size_t BSE = (size_t)kB * kS * kE;           // 4M floats = 16 MB
  float* ws = (float*)d_ws;                          // needs 64 MB
  float* Qp = ws;
  float* Kp = Qp + BSE;
  float* Vp = Kp + BSE;
  float* Ah = Vp + BSE;

  const dim3 blk(256);
  const dim3 gproj(kM / 64, kE / 128);               // 64 x 8 blocks

  gemm_bias_wmma<<<gproj, blk, 0, stream>>>(query, Wq, bq, Qp, kM, kE, kE);
  gemm_bias_wmma<<<gproj, blk, 0, stream>>>(key,   Wk, bk, Kp, kM, kE, kE);
  gemm_bias_wmma<<<gproj, blk, 0, stream>>>(value, Wv, bv, Vp, kM, kE, kE);

  const size_t shmem =
      (size_t)(16 * kSStride + 16 * kQStride + (2 * kMaxRel + 1) + 3 + 256 + 32) *
      sizeof(float);  // ~138 KB, fits 320 KB/WGP LDS
  attn_softmax_av<<<dim3(kB * kH * (kS / 16)), blk, shmem, stream>>>(
      Qp, Kp, Vp, tbl, attn, Ah);

  gemm_bias_wmma<<<gproj, blk, 0, stream>>>(Ah, Wo, bo, out, kM, kE, kE);
}